// DEBlock_48945447305226
// MI455X (gfx1250) — compile-verified
//
#include <hip/hip_runtime.h>
#include <math.h>
#include <stdint.h>

typedef float v2f __attribute__((ext_vector_type(2)));
typedef float v8f __attribute__((ext_vector_type(8)));

#define B_   4
#define C_   128
#define N_   8192
#define K_   16
#define H_   64
#define NDIR 32
#define FSZ  (B_ * C_ * N_)   // 4,194,304 floats = 16 MB per region
#define PTS  4

__device__ __forceinline__ v8f wmma4(v2f a, v2f b, v8f c) {
  // V_WMMA_F32_16X16X4_F32: D(16x16,f32) = A(16x4,f32) * B(4x16,f32) + C
  return __builtin_amdgcn_wmma_f32_16x16x4_f32(false, a, false, b, (short)0, c,
                                               false, false);
}

__device__ __forceinline__ float gelu_exact(float x) {
  return 0.5f * x * (1.0f + erff(x * 0.70710678118654752f));
}

// CDNA5 async copy global -> LDS (ASYNCcnt-tracked), 16B per lane.
__device__ __forceinline__ void async_load_b128(unsigned lds_addr,
                                                const void* gaddr) {
  asm volatile("global_load_async_to_lds_b128 %0, %1, off"
               :: "v"(lds_addr), "v"(gaddr) : "memory");
}
__device__ __forceinline__ void wait_async0() {
  asm volatile("s_wait_asynccnt 0x0" ::: "memory");
}

// Low 32 bits of a flat shared-memory address == LDS byte offset (ISA 10.2).
__device__ __forceinline__ unsigned lds_off(const void* p) {
  return (unsigned)(uintptr_t)p;
}

// ---------------------------------------------------------------------------
// Kernel A: f1/f2/f3 = l_wi(128x128) * f(b) + l_bi     (fp32 WMMA)
// grid = (N/64, B, 3), block = 256 (8 waves).
// The shared 128x64 activation tile is async-staged into LDS once per WG;
// all 8 waves read B fragments from the DS pipe.
// ---------------------------------------------------------------------------
__global__ __launch_bounds__(256)
void gemm3_kernel(const float* __restrict__ f,
                  const float* __restrict__ w1, const float* __restrict__ b1,
                  const float* __restrict__ w2, const float* __restrict__ b2,
                  const float* __restrict__ w3, const float* __restrict__ b3,
                  float* __restrict__ o1, float* __restrict__ o2,
                  float* __restrict__ o3) {
  __shared__ float Xs[C_ * 64];  // 32 KB tile: rows k=0..127, cols n0..n0+63

  const float* W; const float* bias; float* Out;
  if (blockIdx.z == 0)      { W = w1; bias = b1; Out = o1; }
  else if (blockIdx.z == 1) { W = w2; bias = b2; Out = o2; }
  else                      { W = w3; bias = b3; Out = o3; }

  const int b = blockIdx.y;
  const float* X = f + (size_t)b * C_ * N_;
  Out += (size_t)b * C_ * N_;

  const int n0   = blockIdx.x * 64;
  const int wave = threadIdx.x >> 5;
  const int lane = threadIdx.x & 31;
  const int half = lane >> 4;
  const int lm   = lane & 15;
  const int m0   = wave * 16;

  // ---- async stage: 2048 x 16B chunks, 8 per thread, coalesced rows ----
  {
    const unsigned base = lds_off(&Xs[0]);
    const int t = threadIdx.x;
#pragma unroll
    for (int i = 0; i < 8; ++i) {
      int q   = t + i * 256;       // chunk id 0..2047
      int row = q >> 4;            // k row 0..127
      int c4  = (q & 15) * 4;      // col offset (floats)
      async_load_b128(base + (unsigned)q * 16,
                      X + (size_t)row * N_ + n0 + c4);
    }
  }
  wait_async0();
  __syncthreads();

  v8f acc[4] = {};
  for (int k0 = 0; k0 < C_; k0 += 4) {
    // A fragment: W[m0+lm][k0+2*half .. +1]  (8B aligned: k0%4==0)
    v2f a = *(const v2f*)(W + (size_t)(m0 + lm) * C_ + k0 + 2 * half);
    const float* x0 = &Xs[(k0 + 2 * half) * 64];
#pragma unroll
    for (int j = 0; j < 4; ++j) {
      int col = j * 16 + lm;
      v2f bf;
      bf.x = x0[col];
      bf.y = x0[64 + col];
      acc[j] = wmma4(a, bf, acc[j]);
    }
  }
#pragma unroll
  for (int j = 0; j < 4; ++j) {
    int col = n0 + j * 16 + lm;
#pragma unroll
    for (int r = 0; r < 8; ++r) {
      int row = m0 + r + 8 * half;
      Out[(size_t)row * N_ + col] = acc[j][r] + bias[row];
    }
  }
}

// ---------------------------------------------------------------------------
// Kernel B: fused directional encoding + KNN max-gather + BN + residual.
// grid = (N/PTS, B), block = 128 (one thread per channel).
// Produces f_mid = f + BN(agg1+agg2+f3) + pe.
// ---------------------------------------------------------------------------
__global__ __launch_bounds__(128)
void gather_pe_kernel(const float* __restrict__ f, const float* __restrict__ dp,
                      const int* __restrict__ qidx,
                      const float* __restrict__ dirv,
                      const float* __restrict__ de_w1,
                      const float* __restrict__ de_g, const float* __restrict__ de_b,
                      const float* __restrict__ de_m, const float* __restrict__ de_v,
                      const float* __restrict__ de_w2, const float* __restrict__ de_b2,
                      const float* __restrict__ lg, const float* __restrict__ lb,
                      const float* __restrict__ lmn, const float* __restrict__ lv,
                      const float* __restrict__ f1, const float* __restrict__ f2,
                      const float* __restrict__ f3, float* __restrict__ fmid) {
  __shared__ float s_vn[NDIR][3];
  __shared__ float s_dpn[PTS][K_][3];
  __shared__ float s_tm[PTS][NDIR];
  __shared__ float s_h[PTS][H_];
  __shared__ int   s_idx[PTS][K_];

  const int b     = blockIdx.y;
  const int nbase = blockIdx.x * PTS;
  const int tid   = threadIdx.x;

  if (tid < NDIR) {
    float x = dirv[tid * 3 + 0], y = dirv[tid * 3 + 1], z = dirv[tid * 3 + 2];
    float inv = 1.0f / fmaxf(sqrtf(x * x + y * y + z * z), 1e-12f);
    s_vn[tid][0] = x * inv; s_vn[tid][1] = y * inv; s_vn[tid][2] = z * inv;
  }
  if (tid < 64) {
    int p = tid >> 4, k = tid & 15;
    int n = nbase + p;
    size_t base = ((size_t)(b * 3 + 0) * N_ + n) * K_ + k;
    float x = dp[base];
    float y = dp[base + (size_t)N_ * K_];
    float z = dp[base + 2 * (size_t)N_ * K_];
    float inv = 1.0f / fmaxf(sqrtf(x * x + y * y + z * z), 1e-12f);
    s_dpn[p][k][0] = x * inv; s_dpn[p][k][1] = y * inv; s_dpn[p][k][2] = z * inv;
  } else {
    int t = tid - 64;
    int p = t >> 4, k = t & 15;
    int n = nbase + p;
    s_idx[p][k] = qidx[((size_t)b * N_ + n) * K_ + k];
  }
  __syncthreads();

  {  // theta_max: 4 points x 32 dirs = 128 threads
    int p = tid >> 5, m = tid & 31;
    float mx = -3.402823466e38f;
#pragma unroll
    for (int k = 0; k < K_; ++k) {
      float d = s_vn[m][0] * s_dpn[p][k][0] + s_vn[m][1] * s_dpn[p][k][1] +
                s_vn[m][2] * s_dpn[p][k][2];
      mx = fmaxf(mx, d);
    }
    s_tm[p][m] = mx;
  }
  __syncthreads();

#pragma unroll
  for (int it = 0; it < 2; ++it) {  // 4 points x 64 hidden = 256 = 2 x 128
    int q = tid + it * 128;
    int p = q >> 6, j = q & 63;
    float acc = 0.0f;
    for (int d = 0; d < NDIR; ++d) acc += de_w1[j * NDIR + d] * s_tm[p][d];
    float bn = (acc - de_m[j]) * rsqrtf(de_v[j] + 1e-5f) * de_g[j] + de_b[j];
    s_h[p][j] = gelu_exact(bn);
  }
  __syncthreads();

  const int c = tid;
  float scale = lg[c] * rsqrtf(lv[c] + 1e-5f);
  float shift = lb[c] - lmn[c] * scale;
  const float* f1c = f1 + ((size_t)b * C_ + c) * N_;
  const float* f2c = f2 + ((size_t)b * C_ + c) * N_;
  const float* f3c = f3 + ((size_t)b * C_ + c) * N_;
  const float* fc  = f  + ((size_t)b * C_ + c) * N_;
  float* fm = fmid + ((size_t)b * C_ + c) * N_;
  float b2c = de_b2[c];

  for (int p = 0; p < PTS; ++p) {
    int n = nbase + p;
    float pe = b2c;
    for (int j = 0; j < H_; ++j) pe += de_w2[c * H_ + j] * s_h[p][j];
    float a1 = -3.402823466e38f, a2 = -3.402823466e38f;
#pragma unroll
    for (int j = 0; j < K_; ++j) {
      int id = s_idx[p][j];
      a1 = fmaxf(a1, f1c[id]);
      a2 = fmaxf(a2, f2c[id]);
    }
    float v  = a1 + (a2 + f2c[n]) + f3c[n];
    float bn = v * scale + shift;
    fm[n] = fc[n] + bn + pe;
  }
}

// ---------------------------------------------------------------------------
// Kernel C: fused 2-stage WMMA MLP, no global h2 buffer.
// grid = (N/64, B), block = 256 (8 waves).
// f_mid tile (128x64) async-staged to LDS; serves stage-1 B operand AND the
// stage-2 residual. Stage 1: Hs(256x64)=gelu(BN(m_w1*Xs)) in 64 KB LDS.
// Stage 2: out = Xs + m_w2 * Hs.  Total LDS = 96 KB.
// ---------------------------------------------------------------------------
__global__ __launch_bounds__(256)
void mlp_kernel(const float* __restrict__ fmid, const float* __restrict__ mw1,
                const float* __restrict__ mg, const float* __restrict__ mb,
                const float* __restrict__ mm, const float* __restrict__ mv,
                const float* __restrict__ mw2, float* __restrict__ out) {
  __shared__ float Hs[256 * 64];  // 64 KB
  __shared__ float Xs[C_ * 64];   // 32 KB f_mid tile

  const int b  = blockIdx.y;
  const int n0 = blockIdx.x * 64;
  const float* Fm = fmid + (size_t)b * C_ * N_;

  const int wave = threadIdx.x >> 5;
  const int lane = threadIdx.x & 31;
  const int half = lane >> 4;
  const int lm   = lane & 15;

  // ---- async stage the f_mid tile ----
  {
    const unsigned base = lds_off(&Xs[0]);
    const int t = threadIdx.x;
#pragma unroll
    for (int i = 0; i < 8; ++i) {
      int q   = t + i * 256;
      int row = q >> 4;
      int c4  = (q & 15) * 4;
      async_load_b128(base + (unsigned)q * 16,
                      Fm + (size_t)row * N_ + n0 + c4);
    }
  }
  wait_async0();
  __syncthreads();

  {  // Stage 1: each wave does rows [16w, 16w+15] and [16w+128, 16w+143]
    v8f acc[2][4] = {};
    for (int k0 = 0; k0 < 128; k0 += 4) {
      const float* x0 = &Xs[(k0 + 2 * half) * 64];
      v2f bf[4];
#pragma unroll
      for (int j = 0; j < 4; ++j) {
        int col = j * 16 + lm;
        bf[j].x = x0[col];
        bf[j].y = x0[64 + col];
      }
#pragma unroll
      for (int t = 0; t < 2; ++t) {
        int row = wave * 16 + t * 128 + lm;
        v2f a = *(const v2f*)(mw1 + (size_t)row * 128 + k0 + 2 * half);
#pragma unroll
        for (int j = 0; j < 4; ++j) acc[t][j] = wmma4(a, bf[j], acc[t][j]);
      }
    }
#pragma unroll
    for (int t = 0; t < 2; ++t) {
#pragma unroll
      for (int r = 0; r < 8; ++r) {
        int row = wave * 16 + t * 128 + r + 8 * half;
        float scale = mg[row] * rsqrtf(mv[row] + 1e-5f);
        float shift = mb[row] - mm[row] * scale;
#pragma unroll
        for (int j = 0; j < 4; ++j) {
          int col = j * 16 + lm;
          Hs[row * 64 + col] = gelu_exact(acc[t][j][r] * scale + shift);
        }
      }
    }
  }
  __syncthreads();

  {  // Stage 2: K=256 from LDS, residual from Xs
    v8f acc[4] = {};
    const int m0 = wave * 16;
    for (int k0 = 0; k0 < 256; k0 += 4) {
      v2f a = *(const v2f*)(mw2 + (size_t)(m0 + lm) * 256 + k0 + 2 * half);
#pragma unroll
      for (int j = 0; j < 4; ++j) {
        int col = j * 16 + lm;
        v2f bf;
        bf.x = Hs[(k0 + 2 * half) * 64 + col];
        bf.y = Hs[(k0 + 2 * half + 1) * 64 + col];
        acc[j] = wmma4(a, bf, acc[j]);
      }
    }
    float* O = out + (size_t)b * C_ * N_;
#pragma unroll
    for (int j = 0; j < 4; ++j) {
      int lcol = j * 16 + lm;
      int col  = n0 + lcol;
#pragma unroll
      for (int r = 0; r < 8; ++r) {
        int row = m0 + r + 8 * half;
        O[(size_t)row * N_ + col] = Xs[row * 64 + lcol] + acc[j][r];
      }
    }
  }
}

// ---------------------------------------------------------------------------
extern "C" void kernel_launch(void* const* d_in, const int* in_sizes, int n_in,
                              void* d_out, int out_size, void* d_ws,
                              size_t ws_size, hipStream_t stream) {
  const float* f     = (const float*)d_in[0];
  const float* dp    = (const float*)d_in[1];
  const int*   qidx  = (const int*)  d_in[2];
  const float* dirv  = (const float*)d_in[3];
  const float* de_w1 = (const float*)d_in[4];
  const float* de_g  = (const float*)d_in[5];
  const float* de_b  = (const float*)d_in[6];
  const float* de_m  = (const float*)d_in[7];
  const float* de_v  = (const float*)d_in[8];
  const float* de_w2 = (const float*)d_in[9];
  const float* de_b2 = (const float*)d_in[10];
  const float* l_w1  = (const float*)d_in[11];
  const float* l_b1  = (const float*)d_in[12];
  const float* l_w2  = (const float*)d_in[13];
  const float* l_b2  = (const float*)d_in[14];
  const float* l_w3  = (const float*)d_in[15];
  const float* l_b3  = (const float*)d_in[16];
  const float* l_g   = (const float*)d_in[17];
  const float* l_bb  = (const float*)d_in[18];
  const float* l_m   = (const float*)d_in[19];
  const float* l_v   = (const float*)d_in[20];
  const float* m_w1  = (const float*)d_in[21];
  const float* m_g   = (const float*)d_in[22];
  const float* m_b   = (const float*)d_in[23];
  const float* m_m   = (const float*)d_in[24];
  const float* m_v   = (const float*)d_in[25];
  const float* m_w2  = (const float*)d_in[26];

  float* f1   = (float*)d_ws;
  float* f2   = f1 + FSZ;
  float* f3   = f2 + FSZ;
  float* fmid = f3 + FSZ;   // total 64 MB of workspace

  gemm3_kernel<<<dim3(N_ / 64, B_, 3), 256, 0, stream>>>(
      f, l_w1, l_b1, l_w2, l_b2, l_w3, l_b3, f1, f2, f3);

  gather_pe_kernel<<<dim3(N_ / PTS, B_), 128, 0, stream>>>(
      f, dp, qidx, dirv, de_w1, de_g, de_b, de_m, de_v, de_w2, de_b2,
      l_g, l_bb, l_m, l_v, f1, f2, f3, fmid);

  mlp_kernel<<<dim3(N_ / 64, B_), 256, 0, stream>>>(
      fmid, m_w1, m_g, m_b, m_m, m_v, m_w2, (float*)d_out);
}